// EmbryoNeuralODE_38560216384171
// MI455X (gfx1250) — compile-verified
//
#include <hip/hip_runtime.h>

typedef __attribute__((ext_vector_type(2))) float v2f;
typedef __attribute__((ext_vector_type(8))) float v8f;

#define N_CELLS 2048
#define N_BATCH 16
#define HID     64
#define CPB     256          // cells per block (8 waves x 2 tiles x 16 rows)
#define AS      68           // LDS activation row stride in floats (conflict-free)
#define EPS_K   1e-8f
#define VOLC    0.1f

static __device__ __forceinline__ v8f wmma4(v2f a, v2f b, v8f c) {
    // D(16x16,f32) = A(16x4,f32) x B(4x16,f32) + C
    return __builtin_amdgcn_wmma_f32_16x16x4_f32(
        /*neg_a=*/false, a, /*neg_b=*/false, b,
        /*c_mod=*/(short)0, c, /*reuse_a=*/false, /*reuse_b=*/false);
}

// Branchless padded load: index is clamped (always in-bounds), pad selects 0
// on the loaded VALUE -> v_cndmask, no exec-mask branching.
static __device__ __forceinline__ float padload(const float* __restrict__ p,
                                                int idx, bool valid) {
    float v = p[valid ? idx : 0];      // clamped, unconditional load
    return valid ? v : 0.0f;
}

// Branchless tanh: 1 - 2/(e^{2x}+1) using v_exp_f32 (2^x) + v_rcp_f32.
// Saturates correctly (exp2->inf => 1, exp2->0 => -1), no exec divergence.
static __device__ __forceinline__ float fast_tanh(float x) {
    const float e2x = __builtin_amdgcn_exp2f(x * 2.8853900817779268f); // e^(2x)
    return 1.0f - 2.0f * __builtin_amdgcn_rcpf(e2x + 1.0f);
}

__launch_bounds__(256)
__global__ void embryo_ode_kernel(
    const float* __restrict__ state,
    const float* __restrict__ W1, const float* __restrict__ b1,
    const float* __restrict__ W2, const float* __restrict__ b2,
    const float* __restrict__ W3, const float* __restrict__ b3,
    const float* __restrict__ S1, const float* __restrict__ bs1,
    const float* __restrict__ S2, const float* __restrict__ bs2,
    float* __restrict__ out)
{
    __shared__ float sx[N_CELLS * 3];       // batch positions       24576 B
    __shared__ float sact[8 * 16 * AS];     // per-wave activations  34816 B
    __shared__ float sinter[CPB * 4];       // interaction vectors    4096 B
    __shared__ float sred[8 * 4];           // COM wave partials
    __shared__ float scom[4];               // COM

    const int t    = threadIdx.x;
    const int lane = t & 31;
    const int wv   = t >> 5;
    const int hi   = lane >> 4;     // lane group (0: lanes 0-15, 1: lanes 16-31)
    const int lm   = lane & 15;     // A row m / B-C column n within tile
    const int kA   = hi * 2;        // A/B VGPR0 K index within 4-slice

    const int b     = blockIdx.x >> 3;   // batch
    const int chunk = blockIdx.x & 7;
    const int c0    = chunk * CPB;       // batch-local first cell of this block
    const float* xb = state + b * (N_CELLS * 3);

    // ---- stage whole batch's positions into LDS ----
    for (int i = t; i < N_CELLS * 3; i += 256) sx[i] = xb[i];
    __syncthreads();

    // ---- center of mass (computed redundantly per block, it's tiny) ----
    {
        float cx = 0.f, cy = 0.f, cz = 0.f;
        for (int j = t; j < N_CELLS; j += 256) {
            cx += sx[j*3+0]; cy += sx[j*3+1]; cz += sx[j*3+2];
        }
        #pragma unroll
        for (int off = 16; off; off >>= 1) {
            cx += __shfl_down(cx, off);
            cy += __shfl_down(cy, off);
            cz += __shfl_down(cz, off);
        }
        if (lane == 0) { sred[wv*4+0]=cx; sred[wv*4+1]=cy; sred[wv*4+2]=cz; }
    }

    // ---- O(N^2) pairwise interaction for this block's 256 cells ----
    {
        const int ci = c0 + t;
        const float xi = sx[ci*3+0], yi = sx[ci*3+1], zi = sx[ci*3+2];
        float ax = 0.f, ay = 0.f, az = 0.f;
        #pragma unroll 4
        for (int j = 0; j < N_CELLS; ++j) {   // uniform LDS address -> broadcast
            const float dx = xi - sx[j*3+0];
            const float dy = yi - sx[j*3+1];
            const float dz = zi - sx[j*3+2];
            const float d2 = dx*dx + dy*dy + dz*dz;
            // fast v_sqrt + v_rcp; j==i gives 0 * rcp(eps) = 0
            const float inv = __builtin_amdgcn_rcpf(__builtin_amdgcn_sqrtf(d2) + EPS_K);
            ax += dx * inv; ay += dy * inv; az += dz * inv;
        }
        sinter[t*4+0] = ax; sinter[t*4+1] = ay; sinter[t*4+2] = az;
    }
    __syncthreads();
    if (t == 0) {
        float ax=0.f, ay=0.f, az=0.f;
        #pragma unroll
        for (int w = 0; w < 8; ++w) { ax+=sred[w*4+0]; ay+=sred[w*4+1]; az+=sred[w*4+2]; }
        const float inv = 1.0f / (float)N_CELLS;
        scom[0]=ax*inv; scom[1]=ay*inv; scom[2]=az*inv;
    }
    __syncthreads();

    // ---- per-wave WMMA MLP pipeline: 2 tiles of 16 cells each ----
    float* act = sact + wv * 16 * AS;   // private 16x64 activation buffer
    const int n3 = lm;                  // output column (valid < 3)
    const bool nv = (n3 < 3);
    const int n3c = nv ? n3 : 0;        // clamped output column

    for (int rep = 0; rep < 2; ++rep) {
        const int tb = (wv + 8*rep) * 16;   // block-local tile base cell
        const int cl = c0 + tb;             // batch-local tile base cell

        // ===== layer 1: x(16x3) @ W1(3x64) + b1 -> tanh -> act =====
        {
            const int rc = cl + lm;         // batch-local row cell
            const int k1 = kA + 1;          // 1 (lanes<16) or 3 (lanes>=16)
            const bool k1v = (k1 < 3);
            v2f a;
            a.x = sx[rc*3 + kA];                            // k = 0 or 2 (<3)
            {
                float v = sx[rc*3 + (k1v ? k1 : 0)];
                a.y = k1v ? v : 0.0f;
            }
            v8f h[4];
            #pragma unroll
            for (int nt = 0; nt < 4; ++nt) {
                const int n = nt*16 + lm;
                v8f c;
                const float bias = b1[n];
                #pragma unroll
                for (int q = 0; q < 8; ++q) c[q] = bias;
                v2f bb;
                bb.x = W1[kA*HID + n];
                bb.y = padload(W1, (k1v ? k1 : 0)*HID + n, k1v);
                h[nt] = wmma4(a, bb, c);
            }
            #pragma unroll
            for (int nt = 0; nt < 4; ++nt)
                #pragma unroll
                for (int q = 0; q < 8; ++q) {
                    const int m = q + 8*hi;
                    act[m*AS + nt*16 + lm] = fast_tanh(h[nt][q]);
                }
        }

        // ===== layer 2: h(16x64) @ W2(64x64) + b2 -> tanh -> act =====
        {
            v8f g[4];
            #pragma unroll
            for (int nt = 0; nt < 4; ++nt) {
                const float bias = b2[nt*16 + lm];
                #pragma unroll
                for (int q = 0; q < 8; ++q) g[nt][q] = bias;
            }
            #pragma unroll 4
            for (int ks = 0; ks < 16; ++ks) {
                const int k0 = ks*4 + kA;
                v2f a;
                a.x = act[lm*AS + k0];
                a.y = act[lm*AS + k0 + 1];
                #pragma unroll
                for (int nt = 0; nt < 4; ++nt) {
                    const int n = nt*16 + lm;
                    v2f bb;
                    bb.x = W2[k0*HID + n];
                    bb.y = W2[(k0+1)*HID + n];
                    g[nt] = wmma4(a, bb, g[nt]);
                }
            }
            #pragma unroll
            for (int nt = 0; nt < 4; ++nt)
                #pragma unroll
                for (int q = 0; q < 8; ++q) {
                    const int m = q + 8*hi;
                    act[m*AS + nt*16 + lm] = fast_tanh(g[nt][q]);
                }
        }

        // ===== layer 3: h2(16x64) @ W3(64x3) + b3 -> individual tile =====
        v8f indiv;
        {
            const float bias = padload(b3, n3c, nv);
            #pragma unroll
            for (int q = 0; q < 8; ++q) indiv[q] = bias;
            #pragma unroll 2
            for (int ks = 0; ks < 16; ++ks) {
                const int k0 = ks*4 + kA;
                v2f a;
                a.x = act[lm*AS + k0];
                a.y = act[lm*AS + k0 + 1];
                v2f bb;
                bb.x = padload(W3, k0*3 + n3c, nv);
                bb.y = padload(W3, (k0+1)*3 + n3c, nv);
                indiv = wmma4(a, bb, indiv);
            }
        }

        // ===== spatial layer 1: cat(x,inter)(16x6) @ S1(6x64) + bs1 -> tanh =====
        {
            v8f e[4];
            #pragma unroll
            for (int nt = 0; nt < 4; ++nt) {
                const float bias = bs1[nt*16 + lm];
                #pragma unroll
                for (int q = 0; q < 8; ++q) e[nt][q] = bias;
            }
            const int lc = tb + lm;   // block-local cell for sinter
            const int rc = cl + lm;   // batch-local cell for sx
            #pragma unroll
            for (int ks = 0; ks < 2; ++ks) {
                const int k0 = ks*4 + kA;      // {0,2,4,6}
                const int k1 = k0 + 1;         // {1,3,5,7}
                // branchless cat[]: both LDS loads unconditional, value select
                v2f a;
                {
                    const float vx = sx[rc*3 + (k0 < 3 ? k0 : 0)];
                    const float vi = sinter[lc*4 + (k0 >= 3 ? (k0 < 6 ? k0-3 : 0) : 0)];
                    a.x = (k0 < 3) ? vx : ((k0 < 6) ? vi : 0.0f);
                }
                {
                    const float vx = sx[rc*3 + (k1 < 3 ? k1 : 0)];
                    const float vi = sinter[lc*4 + (k1 >= 3 ? (k1 < 6 ? k1-3 : 0) : 0)];
                    a.y = (k1 < 3) ? vx : ((k1 < 6) ? vi : 0.0f);
                }
                #pragma unroll
                for (int nt = 0; nt < 4; ++nt) {
                    const int n = nt*16 + lm;
                    v2f bb;
                    bb.x = padload(S1, (k0 < 6 ? k0 : 0)*HID + n, k0 < 6);
                    bb.y = padload(S1, (k1 < 6 ? k1 : 0)*HID + n, k1 < 6);
                    e[nt] = wmma4(a, bb, e[nt]);
                }
            }
            #pragma unroll
            for (int nt = 0; nt < 4; ++nt)
                #pragma unroll
                for (int q = 0; q < 8; ++q) {
                    const int m = q + 8*hi;
                    act[m*AS + nt*16 + lm] = fast_tanh(e[nt][q]);
                }
        }

        // ===== spatial layer 2: (16x64) @ S2(64x3) + bs2 -> eff tile =====
        v8f eff;
        {
            const float bias = padload(bs2, n3c, nv);
            #pragma unroll
            for (int q = 0; q < 8; ++q) eff[q] = bias;
            #pragma unroll 2
            for (int ks = 0; ks < 16; ++ks) {
                const int k0 = ks*4 + kA;
                v2f a;
                a.x = act[lm*AS + k0];
                a.y = act[lm*AS + k0 + 1];
                v2f bb;
                bb.x = padload(S2, k0*3 + n3c, nv);
                bb.y = padload(S2, (k0+1)*3 + n3c, nv);
                eff = wmma4(a, bb, eff);
            }
        }

        // ===== fused output: individual + eff + volume =====
        if (nv) {
            const float comk = scom[n3];
            #pragma unroll
            for (int q = 0; q < 8; ++q) {
                const int m = q + 8*hi;
                const int cellB = cl + m;                 // batch-local cell
                const int gcell = b * N_CELLS + cellB;    // global cell
                const float vol = -VOLC * (sx[cellB*3 + n3] - comk);
                out[gcell*3 + n3] = indiv[q] + eff[q] + vol;
            }
        }
    }
}

extern "C" void kernel_launch(void* const* d_in, const int* in_sizes, int n_in,
                              void* d_out, int out_size, void* d_ws, size_t ws_size,
                              hipStream_t stream) {
    (void)in_sizes; (void)n_in; (void)out_size; (void)d_ws; (void)ws_size;
    // inputs: t, state, W1, b1, W2, b2, W3, b3, S1, bs1, S2, bs2 (all f32)
    const float* state = (const float*)d_in[1];
    const float* W1  = (const float*)d_in[2];
    const float* b1  = (const float*)d_in[3];
    const float* W2  = (const float*)d_in[4];
    const float* b2  = (const float*)d_in[5];
    const float* W3  = (const float*)d_in[6];
    const float* b3  = (const float*)d_in[7];
    const float* S1  = (const float*)d_in[8];
    const float* bs1 = (const float*)d_in[9];
    const float* S2  = (const float*)d_in[10];
    const float* bs2 = (const float*)d_in[11];
    float* out = (float*)d_out;

    const int blocks = N_BATCH * (N_CELLS / CPB);   // 16 * 8 = 128
    embryo_ode_kernel<<<dim3(blocks), dim3(256), 0, stream>>>(
        state, W1, b1, W2, b2, W3, b3, S1, bs1, S2, bs2, out);
}